// DotAttention_50989851738893
// MI455X (gfx1250) — compile-verified
//
#include <hip/hip_runtime.h>
#include <stdint.h>

// ---------------------------------------------------------------------------
// DotAttention on MI455X (gfx1250): single-pass flash-style split softmax,
// software-pipelined with CDNA5 async global->LDS DMA.
//
// inputs[32,4096,1024] fp32 = 512MB dominates (arith intensity ~0.5 FLOP/B,
// HBM-bound; one-pass roofline ~22us @ 23.3TB/s). Each workgroup walks 4
// consecutive 64-row chunks as 8 half-tiles (32x1024 = 128KB), double-
// buffered in LDS: the async-load batch for half h+1 is issued before
// computing half h, so 7/8 of HBM traffic hides behind WMMA/LDS compute.
// Scores use V_WMMA_F32_16X16X4_F32 (exact fp32). Per-64-row-chunk partials
// (m, l, acc[1024]) are built with an online softmax rescale and combined by
// a tiny second kernel.
// ---------------------------------------------------------------------------

#define BATCH   32
#define SEQ     4096
#define DIM     1024
#define CHUNK   64                 // rows per ws partial record
#define CHUNKS  (SEQ / CHUNK)      // 64 records per batch
#define CHUNK_H 32                 // rows per LDS half-tile
#define CPW     4                  // 64-row chunks per workgroup
#define HALVES  (2 * CPW)          // 8 half-tiles per workgroup
#define PITCH   1028               // LDS row pitch in floats (1024+4: bank dodge)
#define WSS     1032               // floats per (b,chunk) record: [m, l, pad.., acc[1024]]
#define NEG_INF_F (-1.0e30f)

typedef __attribute__((ext_vector_type(2))) float v2f;
typedef __attribute__((ext_vector_type(8))) float v8f;
typedef int v4i __attribute__((vector_size(16)));          // matches builtin param
typedef v4i __attribute__((address_space(1))) as1_v4i;
typedef v4i __attribute__((address_space(3))) as3_v4i;

#if defined(__gfx1250__) && __has_builtin(__builtin_amdgcn_wmma_f32_16x16x4_f32)
#define USE_WMMA 1
#else
#define USE_WMMA 0
#endif

#if defined(__gfx1250__) && __has_builtin(__builtin_amdgcn_global_load_async_to_lds_b128)
#define USE_ASYNC 1
#else
#define USE_ASYNC 0
#endif

__device__ __forceinline__ void wait_async_zero() {
#if USE_ASYNC
#if __has_builtin(__builtin_amdgcn_s_wait_asynccnt)
    __builtin_amdgcn_s_wait_asynccnt(0);
#else
    asm volatile("s_wait_asynccnt 0" ::: "memory");
#endif
#endif
}

// ---------------------------------------------------------------------------
// Kernel 1: grid = (CHUNKS/CPW, BATCH) = (16, 32), block = 128 (4 waves).
// ---------------------------------------------------------------------------
__global__ __launch_bounds__(128) void attn_partial_kernel(
    const float* __restrict__ ctx,
    const float* __restrict__ inp,
    const int*   __restrict__ mask,
    float*       __restrict__ ws)
{
    __shared__ float s_in[2][CHUNK_H * PITCH];  // 2 x 128.5KB double buffer
    __shared__ float s_ctx[DIM];
    __shared__ float s_sc2[2 * CHUNK_H];        // per-D-half partial scores
    __shared__ float s_scores[CHUNK_H];
    __shared__ float s_p[CHUNK_H];
    __shared__ float s_e[2];

    const int t    = threadIdx.x;
    const int lane = t & 31;
    const int wave = t >> 5;
    const int b    = blockIdx.y;
    const int cw   = blockIdx.x;
    const int wg_row0 = cw * CPW * CHUNK;

    // stage context vector once
    for (int i = t; i < DIM; i += 128)
        s_ctx[i] = ctx[(size_t)b * DIM + i];

    // issue the async-load batch for half-tile h into dstbuf
    auto issue_half = [&](float* dstbuf, int h) {
        const float* g = inp + ((size_t)b * SEQ + wg_row0 + (size_t)h * CHUNK_H) * DIM;
#if USE_ASYNC
        for (int i = t; i < CHUNK_H * (DIM / 4); i += 128) {
            const int r   = i >> 8;             // DIM/4 == 256 segments per row
            const int seg = i & 255;
            as1_v4i* gp = (as1_v4i*)(uintptr_t)
                (const void*)(g + (size_t)r * DIM + seg * 4);
            as3_v4i* lp = (as3_v4i*)(uintptr_t)
                (uint32_t)(uintptr_t)(void*)&dstbuf[r * PITCH + seg * 4];
            __builtin_amdgcn_global_load_async_to_lds_b128(gp, lp, 0, 0);
        }
#else
        for (int i = t; i < CHUNK_H * (DIM / 4); i += 128) {
            const int r   = i >> 8;
            const int seg = i & 255;
            const float4 v = *(const float4*)(g + (size_t)r * DIM + seg * 4);
            *(float4*)&dstbuf[r * PITCH + seg * 4] = v;
        }
#endif
    };

    issue_half(s_in[0], 0);                     // prologue prefetch

    // online softmax running state; meaningful in wave 0 (lane-uniform there)
    float m_run = -3.0e38f, l_run = 0.f;
    const int d0 = t * 8;
    float4 acc0 = {0.f, 0.f, 0.f, 0.f};
    float4 acc1 = {0.f, 0.f, 0.f, 0.f};

    for (int h = 0; h < HALVES; ++h) {
        float* buf = s_in[h & 1];
        wait_async_zero();                      // own wave's DMA for buf done
        __syncthreads();                        // everyone's DMA + prior reads done

        if (h + 1 < HALVES)
            issue_half(s_in[(h + 1) & 1], h + 1);   // overlap next DMA w/ compute

        // ---- scores: 4 waves = 2 row-halves x 2 D-halves, WMMA f32 16x16x4 ----
#if USE_WMMA
        {
            v8f c = {0.f, 0.f, 0.f, 0.f, 0.f, 0.f, 0.f, 0.f};
            // A layout (ISA 7.12.2): lane%16 = M, lane/16 selects K pair
            // {0,1}/{2,3}; VGPR index increments K -> per-lane float2.
            const int arow  = (wave & 1) * 16 + (lane & 15);
            const int kh    = (lane >> 4) * 2;
            const int kbase = (wave >> 1) * (DIM / 2);
            #pragma unroll 4
            for (int k0 = kbase; k0 < kbase + DIM / 2; k0 += 4) {
                v2f a  = *(const v2f*)&buf[arow * PITCH + k0 + kh];
                v2f bb = *(const v2f*)&s_ctx[k0 + kh];  // broadcast over N
                c = __builtin_amdgcn_wmma_f32_16x16x4_f32(
                        false, a, false, bb, (short)0, c, false, false);
            }
            // C layout: lane<16 -> M=vgpr, lane>=16 -> M=vgpr+8 (all N equal)
            if ((lane & 15) == 0) {
                const int base = (wave >> 1) * CHUNK_H + (wave & 1) * 16
                               + (lane >> 4) * 8;
                #pragma unroll
                for (int i = 0; i < 8; ++i) s_sc2[base + i] = c[i];
            }
        }
#else
        if (t < CHUNK_H) {
            float sc = 0.f;
            for (int d = 0; d < DIM; ++d) sc += s_ctx[d] * buf[t * PITCH + d];
            s_sc2[t] = sc;
            s_sc2[CHUNK_H + t] = 0.f;
        }
#endif
        __syncthreads();

        // ---- combine D-halves + mask ----
        const int row0 = wg_row0 + h * CHUNK_H;
        if (t < CHUNK_H) {
            const float sc = s_sc2[t] + s_sc2[CHUNK_H + t];
            const int mk = mask[(size_t)b * SEQ + row0 + t];
            s_scores[t] = (mk != 0) ? sc : NEG_INF_F;
        }
        __syncthreads();

        // ---- wave 0: half-tile stats + online merge into (m_run, l_run) ----
        if (wave == 0) {
            const float a0 = s_scores[lane];
            float mx = a0;
            #pragma unroll
            for (int off = 16; off > 0; off >>= 1)
                mx = fmaxf(mx, __shfl_xor(mx, off, 32));
            const float m_new = fmaxf(m_run, mx);
            const float e_run = expf(m_run - m_new);
            const float e_h   = expf(mx - m_new);
            const float p0 = expf(a0 - mx) * e_h;   // pre-scaled weights
            s_p[lane] = p0;
            float ss = p0;
            #pragma unroll
            for (int off = 16; off > 0; off >>= 1)
                ss += __shfl_xor(ss, off, 32);
            l_run = l_run * e_run + ss;
            m_run = m_new;
            if (lane == 0) s_e[0] = e_run;
        }
        __syncthreads();

        // ---- weighted accumulation: thread owns 8 columns ----
        const float e_run = s_e[0];
        acc0.x *= e_run; acc0.y *= e_run; acc0.z *= e_run; acc0.w *= e_run;
        acc1.x *= e_run; acc1.y *= e_run; acc1.z *= e_run; acc1.w *= e_run;
        for (int r = 0; r < CHUNK_H; ++r) {
            const float pr = s_p[r];
            const float4 v0 = *(const float4*)&buf[r * PITCH + d0];
            const float4 v1 = *(const float4*)&buf[r * PITCH + d0 + 4];
            acc0.x += pr * v0.x; acc0.y += pr * v0.y;
            acc0.z += pr * v0.z; acc0.w += pr * v0.w;
            acc1.x += pr * v1.x; acc1.y += pr * v1.y;
            acc1.z += pr * v1.z; acc1.w += pr * v1.w;
        }

        // ---- end of a 64-row chunk: emit partial record, reset state ----
        if (h & 1) {
            const int cix = cw * CPW + (h >> 1);
            float* wp = ws + ((size_t)b * CHUNKS + cix) * WSS;
            if (t == 0) { wp[0] = m_run; wp[1] = l_run; }  // t==0 is wave0/lane0
            *(float4*)(wp + 8 + d0)     = acc0;
            *(float4*)(wp + 8 + d0 + 4) = acc1;
            acc0 = {0.f, 0.f, 0.f, 0.f};
            acc1 = {0.f, 0.f, 0.f, 0.f};
            m_run = -3.0e38f;
            l_run = 0.f;
        }
    }
}

// ---------------------------------------------------------------------------
// Kernel 2: combine the 64 chunk partials per batch.
//   out[b,d] = sum_c exp(m_c - M)/L * acc_c[d],  L = sum_c l_c*exp(m_c - M)
// ---------------------------------------------------------------------------
__global__ __launch_bounds__(128) void attn_combine_kernel(
    const float* __restrict__ ws,
    float*       __restrict__ out)
{
    __shared__ float s_m[CHUNKS];
    __shared__ float s_l[CHUNKS];
    __shared__ float s_scale[CHUNKS];

    const int t    = threadIdx.x;
    const int lane = t & 31;
    const int wave = t >> 5;
    const int b    = blockIdx.x;

    if (t < CHUNKS) {
        const float* p = ws + ((size_t)b * CHUNKS + t) * WSS;
        s_m[t] = p[0];
        s_l[t] = p[1];
    }
    __syncthreads();

    if (wave == 0) {
        const float mA = s_m[lane], mB = s_m[lane + 32];
        float M = fmaxf(mA, mB);
        #pragma unroll
        for (int off = 16; off > 0; off >>= 1)
            M = fmaxf(M, __shfl_xor(M, off, 32));
        const float eA = expf(mA - M);
        const float eB = expf(mB - M);
        float L = s_l[lane] * eA + s_l[lane + 32] * eB;
        #pragma unroll
        for (int off = 16; off > 0; off >>= 1)
            L += __shfl_xor(L, off, 32);
        const float inv = 1.0f / L;   // L >= 1 (each chunk's max row gives e^0)
        s_scale[lane]      = eA * inv;
        s_scale[lane + 32] = eB * inv;
    }
    __syncthreads();

    const int d0 = t * 8;
    float4 acc0 = {0.f, 0.f, 0.f, 0.f};
    float4 acc1 = {0.f, 0.f, 0.f, 0.f};
    for (int c = 0; c < CHUNKS; ++c) {
        const float sc = s_scale[c];
        const float* ap = ws + ((size_t)b * CHUNKS + c) * WSS + 8 + d0;
        const float4 v0 = *(const float4*)ap;
        const float4 v1 = *(const float4*)(ap + 4);
        acc0.x += sc * v0.x; acc0.y += sc * v0.y;
        acc0.z += sc * v0.z; acc0.w += sc * v0.w;
        acc1.x += sc * v1.x; acc1.y += sc * v1.y;
        acc1.z += sc * v1.z; acc1.w += sc * v1.w;
    }
    *(float4*)(out + (size_t)b * DIM + d0)     = acc0;
    *(float4*)(out + (size_t)b * DIM + d0 + 4) = acc1;
}

// ---------------------------------------------------------------------------
extern "C" void kernel_launch(void* const* d_in, const int* in_sizes, int n_in,
                              void* d_out, int out_size, void* d_ws, size_t ws_size,
                              hipStream_t stream) {
    const float* ctx  = (const float*)d_in[0];   // [32, 1, 1024]
    const float* inp  = (const float*)d_in[1];   // [32, 4096, 1024]
    const int*   mask = (const int*)d_in[2];     // [32, 4096]
    float* out = (float*)d_out;                  // [32, 1024]
    float* ws  = (float*)d_ws;                   // 32*64*1032*4 B ~= 8.5 MB

    dim3 g1(CHUNKS / CPW, BATCH);                // (16, 32), 512 WGs
    attn_partial_kernel<<<g1, 128, 0, stream>>>(ctx, inp, mask, ws);
    attn_combine_kernel<<<dim3(BATCH), 128, 0, stream>>>(ws, out);
}